// NearestEmbedEMA_45999099740650
// MI455X (gfx1250) — compile-verified
//
#include <hip/hip_runtime.h>
#include <hip/hip_bf16.h>
#include <stddef.h>
#include <stdint.h>

// ---------------------------------------------------------------------------
// VQ-VAE NearestEmbedEMA for MI455X (gfx1250, wave32).
//   x:           (32, 64, 64, 64) f32   -> N=131072 tokens, D=64
//   weight:      (64, 512) f32          -> 512 codewords
// Outputs (concatenated): result(8388608) | argmin(131072,i32) |
//                         new_weight(32768) | new_cluster_size(512) |
//                         new_embed_avg(32768)
//
// Matrix op: V_WMMA_F32_16X16X4_F32 (exact fp32 — kernel is HBM-bound, no
// reason to drop precision).  Codebook staged L2 -> LDS with
// GLOBAL_LOAD_ASYNC_TO_LDS_B128 (ASYNCcnt).
// ---------------------------------------------------------------------------

#define N_EMB   512
#define EMB_DIM 64
#define HWSZ    4096          // 64*64
#define NTOK    131072        // 32*4096
#define NRES    8388608       // 32*64*4096
#define DECAY   0.99f
#define EPSV    1e-5f

typedef __attribute__((ext_vector_type(2))) float v2f;
typedef __attribute__((ext_vector_type(8))) float v8f;

// ---------------------------------------------------------------------------
// Kernel 1: codeword squared norms -> ws; zero the count / embed_sum slots.
// ---------------------------------------------------------------------------
__global__ __launch_bounds__(256) void vq_prep(const float* __restrict__ w,
                                               float* __restrict__ wnorm,
                                               float* __restrict__ gcount,
                                               float* __restrict__ gesum) {
  int i = blockIdx.x * blockDim.x + threadIdx.x;
  if (i < N_EMB) {
    float s = 0.f;
#pragma unroll 8
    for (int d = 0; d < EMB_DIM; ++d) {
      float v = w[d * N_EMB + i];
      s += v * v;
    }
    wnorm[i]  = s;
    gcount[i] = 0.f;
  }
  for (int o = i; o < EMB_DIM * N_EMB; o += blockDim.x * gridDim.x)
    gesum[o] = 0.f;
}

// ---------------------------------------------------------------------------
// Kernel 2: argmin via fp32 WMMA.  One wave handles 16 tokens; the full
// codebook (128KB) + norms live in dynamic LDS shared by the block's 8 waves.
// Codebook is staged with async global->LDS B128 copies (no VGPR round-trip).
// ---------------------------------------------------------------------------
__global__ __launch_bounds__(256) void vq_argmin_wmma(
    const float* __restrict__ x, const float* __restrict__ w,
    const float* __restrict__ wnorm, int* __restrict__ amin) {
  extern __shared__ float lds[];              // [64*512 weight][512 norms]
  float* lw  = lds;
  float* lwn = lds + EMB_DIM * N_EMB;

  // Async-stage the 128KB codebook: GLOBAL_LOAD_ASYNC_TO_LDS_B128, GVS mode:
  //   copies global SADDR+VADDR to the LDS address in VDST  (16B per op, ASYNCcnt).
  // Low 32 bits of a generic __shared__ pointer are the LDS byte address.
  for (int i = threadIdx.x; i < (EMB_DIM * N_EMB) / 4; i += 256) {
    unsigned ldsOff = (unsigned)(size_t)(void*)(lw + 4 * i);
    unsigned gOff   = 16u * (unsigned)i;
    asm volatile("global_load_async_to_lds_b128 %0, %1, %2"
                 :
                 : "v"(ldsOff), "v"(gOff), "s"(w)
                 : "memory");
  }
  for (int i = threadIdx.x; i < N_EMB; i += 256) lwn[i] = wnorm[i];
  asm volatile("s_wait_asynccnt 0x0" ::: "memory");
  __syncthreads();

  const int lane = threadIdx.x & 31;
  const int wave = threadIdx.x >> 5;
  const int t0   = (blockIdx.x * 8 + wave) * 16;   // wave's first token
  const int half = lane >> 4;                      // 0: rows 0-7 / K even pair
  const int col  = lane & 15;

  // A fragment layout for V_WMMA_F32_16X16X4_F32 (16x4, 2 VGPRs):
  //   VGPR0: lanes0-15 K=k+0 (M=lane), lanes16-31 K=k+2 (M=lane-16)
  //   VGPR1: lanes0-15 K=k+1,          lanes16-31 K=k+3
  const int bIdx = t0 >> 12;
  const int hw   = (t0 & (HWSZ - 1)) + col;        // token row M = col
  const float* xb = x + (size_t)bIdx * EMB_DIM * HWSZ;

  v2f a[16];
#pragma unroll
  for (int s = 0; s < 16; ++s) {
    int d0 = 4 * s + 2 * half;
    a[s].x = xb[(size_t)(d0)     * HWSZ + hw];
    a[s].y = xb[(size_t)(d0 + 1) * HWSZ + hw];
  }

  float bestV[8];
  int   bestI[8];
#pragma unroll
  for (int r = 0; r < 8; ++r) { bestV[r] = 3.4e38f; bestI[r] = 0; }

  for (int j0 = 0; j0 < N_EMB; j0 += 16) {
    v8f c0 = {0.f, 0.f, 0.f, 0.f, 0.f, 0.f, 0.f, 0.f};
    v8f c1 = {0.f, 0.f, 0.f, 0.f, 0.f, 0.f, 0.f, 0.f};
#pragma unroll
    for (int s = 0; s < 16; s += 2) {              // two accumulator chains
      int k0 = 4 * s + 2 * half;
      v2f b0, b1;
      b0.x = lw[(k0)     * N_EMB + j0 + col];
      b0.y = lw[(k0 + 1) * N_EMB + j0 + col];
      b1.x = lw[(k0 + 4) * N_EMB + j0 + col];
      b1.y = lw[(k0 + 5) * N_EMB + j0 + col];
      c0 = __builtin_amdgcn_wmma_f32_16x16x4_f32(false, a[s],     false, b0,
                                                 (short)0, c0, false, false);
      c1 = __builtin_amdgcn_wmma_f32_16x16x4_f32(false, a[s + 1], false, b1,
                                                 (short)0, c1, false, false);
    }
    float wn  = lwn[j0 + col];
    int   idx = j0 + col;
#pragma unroll
    for (int r = 0; r < 8; ++r) {
      // dist = ||x||^2 - 2 x.w + ||w||^2 ; row term constant -> drop for argmin
      float score = wn - 2.0f * (c0[r] + c1[r]);
      if (score < bestV[r]) { bestV[r] = score; bestI[r] = idx; }  // j0 ascends
    }
  }

  // (value,index) min-reduce across the 16 lanes of each half-wave.
  // Tie-break: lowest index (JAX argmin semantics).
#pragma unroll
  for (int m = 1; m < 16; m <<= 1) {
#pragma unroll
    for (int r = 0; r < 8; ++r) {
      float ov = __shfl_xor(bestV[r], m, 32);
      int   oi = __shfl_xor(bestI[r], m, 32);
      if (ov < bestV[r] || (ov == bestV[r] && oi < bestI[r])) {
        bestV[r] = ov;
        bestI[r] = oi;
      }
    }
  }
  if (col == 0) {
#pragma unroll
    for (int r = 0; r < 8; ++r)
      amin[t0 + half * 8 + r] = bestI[r];   // lane0 -> rows 0-7, lane16 -> 8-15
  }
}

// ---------------------------------------------------------------------------
// Kernel 3: result[b,d,h,w] = weight[d, argmin[b,h,w]]  (coalesced writes)
// ---------------------------------------------------------------------------
__global__ __launch_bounds__(256) void vq_gather(const float* __restrict__ w,
                                                 const int* __restrict__ amin,
                                                 float* __restrict__ out) {
  int i = blockIdx.x * blockDim.x + threadIdx.x;
  if (i >= NRES) return;
  int hw = i & (HWSZ - 1);
  int d  = (i >> 12) & (EMB_DIM - 1);
  int b  = i >> 18;
  int t  = (b << 12) | hw;
  out[i] = w[d * N_EMB + amin[t]];
}

// ---------------------------------------------------------------------------
// Kernel 4: segment sums (counts + per-feature sums) via LDS float atomics,
// flushed with sparse global float atomics.
// ---------------------------------------------------------------------------
__global__ __launch_bounds__(256) void vq_segsum(const float* __restrict__ x,
                                                 const int* __restrict__ amin,
                                                 float* __restrict__ gcount,
                                                 float* __restrict__ gesum) {
  extern __shared__ float lds[];            // [512 counts][64*512 embed_sum]
  float* lcnt = lds;
  float* les  = lds + N_EMB;
  for (int i = threadIdx.x; i < N_EMB + EMB_DIM * N_EMB; i += blockDim.x)
    lds[i] = 0.f;
  __syncthreads();

  for (int t = blockIdx.x * blockDim.x + threadIdx.x; t < NTOK;
       t += gridDim.x * blockDim.x) {
    int idx = amin[t];
    int b = t >> 12, hw = t & (HWSZ - 1);
    const float* xb = x + (size_t)b * EMB_DIM * HWSZ + hw;
    atomicAdd(&lcnt[idx], 1.0f);
#pragma unroll 8
    for (int d = 0; d < EMB_DIM; ++d)
      atomicAdd(&les[d * N_EMB + idx], xb[(size_t)d * HWSZ]);
  }
  __syncthreads();

  for (int i = threadIdx.x; i < N_EMB; i += blockDim.x)
    if (lcnt[i] != 0.f) atomicAdd(&gcount[i], lcnt[i]);
  for (int i = threadIdx.x; i < EMB_DIM * N_EMB; i += blockDim.x)
    if (les[i] != 0.f) atomicAdd(&gesum[i], les[i]);
}

// ---------------------------------------------------------------------------
// Kernel 5: EMA finalize.  ncs slot holds raw counts on entry, nea slot holds
// raw embed_sum on entry; both rewritten in place.
// ---------------------------------------------------------------------------
__global__ __launch_bounds__(512) void vq_finalize(
    const float* __restrict__ cluster_size, const float* __restrict__ embed_avg,
    float* __restrict__ ncs, float* __restrict__ nea, float* __restrict__ nw) {
  __shared__ float red[N_EMB];
  int j = threadIdx.x;
  float cnt = ncs[j];
  if (cnt == 0.f) cnt = 1.f;                       // clamp empty clusters
  float v = DECAY * cluster_size[j] + (1.f - DECAY) * cnt;
  ncs[j] = v;
  red[j] = v;
  __syncthreads();
  for (int s = 256; s > 0; s >>= 1) {
    if (j < s) red[j] += red[j + s];
    __syncthreads();
  }
  float n  = red[0];
  float cs = (v + EPSV) / (n + N_EMB * EPSV) * n;
#pragma unroll 8
  for (int d = 0; d < EMB_DIM; ++d) {
    int o = d * N_EMB + j;
    float e = DECAY * embed_avg[o] + (1.f - DECAY) * nea[o];
    nea[o] = e;                                    // new_embed_avg
    nw[o]  = e / cs;                               // new_weight
  }
}

// ---------------------------------------------------------------------------
extern "C" void kernel_launch(void* const* d_in, const int* in_sizes, int n_in,
                              void* d_out, int out_size, void* d_ws,
                              size_t ws_size, hipStream_t stream) {
  const float* x      = (const float*)d_in[0];   // (32,64,64,64)
  const float* weight = (const float*)d_in[1];   // (64,512)
  const float* cs_in  = (const float*)d_in[2];   // (512,)
  const float* ea_in  = (const float*)d_in[3];   // (64,512)

  float* out      = (float*)d_out;
  float* o_result = out;                                   // 8388608
  int*   o_amin   = (int*)(out + NRES);                    // 131072 (int32 raw)
  float* o_weight = out + NRES + NTOK;                     // 32768
  float* o_cs     = o_weight + EMB_DIM * N_EMB;            // 512
  float* o_eavg   = o_cs + N_EMB;                          // 32768

  float* wnorm = (float*)d_ws;                             // 512 floats

  const size_t lds_bytes =
      (size_t)(EMB_DIM * N_EMB + N_EMB) * sizeof(float);   // ~130 KB / 320 KB

  vq_prep<<<128, 256, 0, stream>>>(weight, wnorm, o_cs, o_eavg);
  vq_argmin_wmma<<<NTOK / 128, 256, lds_bytes, stream>>>(x, weight, wnorm,
                                                         o_amin);
  vq_gather<<<NRES / 256, 256, 0, stream>>>(weight, o_amin, o_result);
  vq_segsum<<<64, 256, lds_bytes, stream>>>(x, o_amin, o_cs, o_eavg);
  vq_finalize<<<1, N_EMB, 0, stream>>>(cs_in, ea_in, o_cs, o_eavg, o_weight);
}